// BatchAllTripletWithClustersLossSemiHard_50946902065286
// MI455X (gfx1250) — compile-verified
//
#include <hip/hip_runtime.h>

#define BB 384
#define DD 512
#define EPSF 1e-8f
#define MARGINF 1.0f
#define NTILE 24              // 384/16 WMMA tiles per dim
#define JBLK 64               // j's per block in quad_loss
#define NJB 6                 // 384/64 j-blocks
#define NPART (BB * NJB)      // partial pairs

typedef __attribute__((ext_vector_type(2))) float v2f;
typedef __attribute__((ext_vector_type(8))) float v8f;

// ---------------------------------------------------------------------------
// Kernel 1: Gram matrix G = X * X^T via V_WMMA_F32_16X16X4_F32.
// One wave32 per 16x16 output tile; 512/4 = 128 WMMA issues per tile.
// A-tile layout (ISA 7.12.2, 32-bit A 16x4): lane<16 -> M=lane, K={0,1};
// lane>=16 -> M=lane-16, K={2,3}. B (4x16) mirrors with N in place of M,
// so for X*X^T both tiles load with identical addressing.
// C/D: VGPR r holds M=r (lanes 0-15) / M=8+r (lanes 16-31), N=lane&15.
// ---------------------------------------------------------------------------
__global__ __launch_bounds__(32) void gram_wmma_f32(const float* __restrict__ X,
                                                    float* __restrict__ G) {
    const int tileJ = blockIdx.x;
    const int tileI = blockIdx.y;
    const int lane  = threadIdx.x & 31;
    const int half  = lane >> 4;
    const int mn    = lane & 15;

    const float* arow = X + (tileI * 16 + mn) * DD + 2 * half;
    const float* brow = X + (tileJ * 16 + mn) * DD + 2 * half;

    v8f c = {};
#pragma unroll 8
    for (int k0 = 0; k0 < DD; k0 += 4) {
        v2f a = *(const v2f*)(arow + k0);
        v2f b = *(const v2f*)(brow + k0);
        c = __builtin_amdgcn_wmma_f32_16x16x4_f32(
                false, a, false, b, (short)0, c, false, false);
    }

#pragma unroll
    for (int r = 0; r < 8; ++r) {
        const int row = tileI * 16 + r + 8 * half;
        const int col = tileJ * 16 + mn;
        G[row * BB + col] = c[r];
    }
}

// ---------------------------------------------------------------------------
// Kernel 2: per-row precompute.
//  sq[i] = G[i][i]; wv[i] = weights[labels[i]]
//  flags[i*BB+k]: bit0 = "first" (leq & rank%2==1), bit1 = "second"
// ---------------------------------------------------------------------------
__global__ __launch_bounds__(BB) void prep_masks(const float* __restrict__ G,
                                                 const int* __restrict__ labels,
                                                 const float* __restrict__ weights,
                                                 float* __restrict__ sq,
                                                 float* __restrict__ wv,
                                                 unsigned char* __restrict__ flags) {
    __shared__ int lab[BB];
    const int i = threadIdx.x;
    lab[i] = labels[i];
    __syncthreads();

    sq[i] = G[i * BB + i];
    const int li = lab[i];
    wv[i] = weights[li];

    int c = 0;
    unsigned char* fi = flags + i * BB;
    for (int k = 0; k < BB; ++k) {
        const bool leq = (lab[k] == li);
        c += leq ? 1 : 0;
        const int rank = c - 1;
        unsigned char f = 0;
        if (leq) f = ((rank & 1) == 1) ? 1u : 2u;  // first : second
        fi[k] = f;
    }
}

// ---------------------------------------------------------------------------
// Kernel 3: fused B^3 reduction. One block per (i, 64-j chunk); 8 waves,
// each wave covers 8 j's. Per-block LDS staging (shared by all j's):
//   dkraw[k]     = d(i,k)
//   dksel[s][k]  = d(i,k), or +INF if blocked (flag bit s set, or k==i)
// Inner loop per k is then pure math: t = relu(d_ij - dksel[k] + 1);
// blocked entries give relu(-INF)=0 and never pass t > EPS/w.
// The lane-varying k==j exclusion is a single post-loop lane-0 correction
// (at k==j the unblocked term is exactly relu(1.0)).
// ---------------------------------------------------------------------------
__global__ __launch_bounds__(256) void quad_loss(const float* __restrict__ G,
                                                 const float* __restrict__ sq,
                                                 const float* __restrict__ wv,
                                                 const unsigned char* __restrict__ flags,
                                                 const int* __restrict__ labels,
                                                 const int* __restrict__ clusters,
                                                 float* __restrict__ partials) {
    const int i     = blockIdx.x;
    const int jBase = blockIdx.y * JBLK;
    const int wave  = threadIdx.x >> 5;
    const int lane  = threadIdx.x & 31;

    __shared__ float dkraw[BB];
    __shared__ float dksel[2][BB];
    __shared__ float ssum[8], scnt[8];

    const float sqi = sq[i];
    const int   li  = labels[i];
    const int   ci  = clusters[i];
    const float* Gi = G + i * BB;
    const unsigned char* Fi = flags + i * BB;
    const float INF = __builtin_huge_valf();

    for (int k = threadIdx.x; k < BB; k += 256) {
        const float dval = fmaxf(sq[k] - 2.0f * Gi[k] + sqi, 0.0f);
        const unsigned char f = Fi[k];
        const bool isI = (k == i);
        dkraw[k]    = dval;
        dksel[0][k] = ((f & 1u) || isI) ? INF : dval;  // ~first  gate
        dksel[1][k] = ((f & 2u) || isI) ? INF : dval;  // ~second gate
    }
    __syncthreads();

    float sum = 0.0f, cnt = 0.0f;

#pragma unroll
    for (int t = 0; t < JBLK / 8; ++t) {
        const int j = jBase + wave + 8 * t;
        if (j == i || labels[j] != li) continue;

        const int   sel = (clusters[j] == ci) ? 0 : 1;
        const float w   = wv[j];
        const float thr = EPSF / w;          // count: w*t > EPS  <=>  t > EPS/w
        const float dij = dkraw[j];
        const float* dsel = dksel[sel];

        float lsum = 0.0f, lcnt = 0.0f;
#pragma unroll
        for (int kk = 0; kk < 3; ++kk) {
            const int k0 = kk * 128 + lane * 4;
            const float4 dv = *(const float4*)(dsel + k0);
            const float t0 = fmaxf(dij - dv.x + MARGINF, 0.0f);
            const float t1 = fmaxf(dij - dv.y + MARGINF, 0.0f);
            const float t2 = fmaxf(dij - dv.z + MARGINF, 0.0f);
            const float t3 = fmaxf(dij - dv.w + MARGINF, 0.0f);
            lsum += t0 + t1 + t2 + t3;
            lcnt += ((t0 > thr) ? 1.0f : 0.0f) + ((t1 > thr) ? 1.0f : 0.0f)
                  + ((t2 > thr) ? 1.0f : 0.0f) + ((t3 > thr) ? 1.0f : 0.0f);
        }
        if (lane == 0) {  // remove the k==j term (exactly relu(1.0) if unblocked)
            const float tj = fmaxf(dij - dsel[j] + MARGINF, 0.0f);
            lsum -= tj;
            if (tj > thr) lcnt -= 1.0f;
        }
        sum += w * lsum;
        cnt += lcnt;
    }

#pragma unroll
    for (int off = 16; off >= 1; off >>= 1) {
        sum += __shfl_xor(sum, off, 32);
        cnt += __shfl_xor(cnt, off, 32);
    }
    if (lane == 0) { ssum[wave] = sum; scnt[wave] = cnt; }
    __syncthreads();
    if (threadIdx.x == 0) {
        float S = 0.0f, C = 0.0f;
#pragma unroll
        for (int wq = 0; wq < 8; ++wq) { S += ssum[wq]; C += scnt[wq]; }
        const int idx = i * gridDim.y + blockIdx.y;
        partials[2 * idx + 0] = S;
        partials[2 * idx + 1] = C;
    }
}

// ---------------------------------------------------------------------------
// Kernel 4: final reduce of NPART partial pairs -> out = sum / (cnt + EPS)
// ---------------------------------------------------------------------------
__global__ __launch_bounds__(256) void finalize(const float* __restrict__ partials,
                                                float* __restrict__ out) {
    float S = 0.0f, C = 0.0f;
    for (int idx = threadIdx.x; idx < NPART; idx += 256) {
        S += partials[2 * idx + 0];
        C += partials[2 * idx + 1];
    }
#pragma unroll
    for (int off = 16; off >= 1; off >>= 1) {
        S += __shfl_xor(S, off, 32);
        C += __shfl_xor(C, off, 32);
    }
    __shared__ float ss[8], sc[8];
    const int wave = threadIdx.x >> 5, lane = threadIdx.x & 31;
    if (lane == 0) { ss[wave] = S; sc[wave] = C; }
    __syncthreads();
    if (threadIdx.x == 0) {
        float s = 0.0f, c = 0.0f;
#pragma unroll
        for (int w = 0; w < 8; ++w) { s += ss[w]; c += sc[w]; }
        out[0] = s / (c + EPSF);
    }
}

// ---------------------------------------------------------------------------
// Launch. Inputs: embeddings f32[384*512], labels i32[384], clusters i32[384],
// weights f32[16]. Output: one f32 scalar.
// Workspace: G (589,824 B) | sq (1.5 KB) | wv (1.5 KB) | partials (18,432 B)
//            | flags (147,456 B)  => ~0.77 MB total.
// ---------------------------------------------------------------------------
extern "C" void kernel_launch(void* const* d_in, const int* in_sizes, int n_in,
                              void* d_out, int out_size, void* d_ws, size_t ws_size,
                              hipStream_t stream) {
    const float* X        = (const float*)d_in[0];
    const int*   labels   = (const int*)d_in[1];
    const int*   clusters = (const int*)d_in[2];
    const float* weights  = (const float*)d_in[3];
    float* out = (float*)d_out;

    float* G        = (float*)d_ws;
    float* sq       = G + BB * BB;
    float* wv       = sq + BB;
    float* partials = wv + BB;
    unsigned char* flags = (unsigned char*)(partials + 2 * NPART);

    gram_wmma_f32<<<dim3(NTILE, NTILE), 32, 0, stream>>>(X, G);
    prep_masks<<<1, BB, 0, stream>>>(G, labels, weights, sq, wv, flags);
    quad_loss<<<dim3(BB, NJB), 256, 0, stream>>>(G, sq, wv, flags, labels, clusters, partials);
    finalize<<<1, 256, 0, stream>>>(partials, out);
}